// LGMLoss_83245056131548
// MI455X (gfx1250) — compile-verified
//
#include <hip/hip_runtime.h>
#include <hip/hip_bf16.h>

// Problem constants (fixed by setup_inputs)
#define B_    2048
#define C_    256
#define F_    256
#define K_    512          // packed K: [x^2 | x] vs [0.5*inv_v | -m*inv_v]
#define ALPHA 0.1f
#define EPSI  1e-8f

#define AST   514          // LDS A row stride (floats): 514 % 64 == 2 -> conflict-free b64 reads
#define MST   257          // LDS md row stride (floats)

typedef __attribute__((ext_vector_type(2))) float v2f;
typedef __attribute__((ext_vector_type(8))) float v8f;

// ---------------------------------------------------------------------------
// Kernel 1: per-class precompute.
//   W[c][k]      = 0.5 * inv_v[c][k]           (k <  256)
//   W[c][256+k]  = -means[c][k] * inv_v[c][k]  (k >= 0)
//   cadd[c]      = 0.5 * sum_f means^2 * inv_v
//   denom[c]     = det(vars_[c])^2 + eps
// Also emits the pass-through outputs (means, vars_).
// ---------------------------------------------------------------------------
__global__ __launch_bounds__(256) void lgm_prep(
    const float* __restrict__ means, const float* __restrict__ vars_,
    float* __restrict__ W, float* __restrict__ cadd, float* __restrict__ denom,
    float* __restrict__ out_means, float* __restrict__ out_vars)
{
    const int c = blockIdx.x;
    const int t = threadIdx.x;

    const float m  = means[c * F_ + t];
    const float v  = vars_[c * F_ + t];
    const float iv = 1.0f / (v + EPSI);

    W[c * K_ + t]       = 0.5f * iv;
    W[c * K_ + F_ + t]  = -m * iv;
    out_means[c * F_ + t] = m;
    out_vars [c * F_ + t] = v;

    __shared__ float ssum[256];
    __shared__ float sprod[256];
    ssum[t]  = m * m * iv;
    sprod[t] = v;
    __syncthreads();
    #pragma unroll
    for (int s = 128; s > 0; s >>= 1) {
        if (t < s) {
            ssum[t]  += ssum[t + s];
            sprod[t] *= sprod[t + s];
        }
        __syncthreads();
    }
    if (t == 0) {
        cadd[c] = 0.5f * ssum[0];
        const float det = sprod[0];
        denom[c] = det * det + EPSI;
    }
}

// ---------------------------------------------------------------------------
// Kernel 2: fused GEMM (V_WMMA_F32_16X16X4_F32) + per-row loss finalize.
// One workgroup = 8 waves = 16 output rows x 256 classes.
// ---------------------------------------------------------------------------
__global__ __launch_bounds__(256) void lgm_gemm(
    const float* __restrict__ feat, const int* __restrict__ labels,
    const float* __restrict__ W, const float* __restrict__ cadd,
    const float* __restrict__ denom, float* __restrict__ loss)
{
    __shared__ float Alds[16 * AST];    // 16 rows x (x^2 | x), padded
    __shared__ float mdlds[16 * MST];   // 16 rows x 256 m_distances, padded

    const int tid = threadIdx.x;
    const int r0  = blockIdx.x * 16;

    // Stage A strip: Alds[r][f] = x^2, Alds[r][256+f] = x
    for (int i = tid; i < 16 * F_; i += 256) {
        const int r = i >> 8;
        const int f = i & 255;
        const float x = feat[(r0 + r) * F_ + f];
        Alds[r * AST + f]       = x * x;
        Alds[r * AST + F_ + f]  = x;
    }
    __syncthreads();

    const int wave = tid >> 5;
    const int lane = tid & 31;

    // WMMA f32 16x16x4 fragment addressing (per ISA layout tables):
    //   A: lane -> M = lane&15, k-pair base = 2*(lane>=16)
    //   B: lane -> N = lane&15, same k-pair base
    const int m  = lane & 15;
    const int kb = (lane >> 4) << 1;           // 0 or 2
    const int n0 = wave * 32 + (lane & 15);    // tile 0 column
    const int n1 = n0 + 16;                    // tile 1 column

    const float* __restrict__ Wp0 = W + (size_t)n0 * K_ + kb;
    const float* __restrict__ Wp1 = W + (size_t)n1 * K_ + kb;
    const float* Ap = Alds + m * AST + kb;

    v8f acc0 = {0.f, 0.f, 0.f, 0.f, 0.f, 0.f, 0.f, 0.f};
    v8f acc1 = {0.f, 0.f, 0.f, 0.f, 0.f, 0.f, 0.f, 0.f};

    #pragma unroll 8
    for (int kk = 0; kk < K_; kk += 4) {
        if ((kk & 31) == 0) {
            __builtin_prefetch(Wp0 + kk + 128, 0, 0);  // global_prefetch_b8
            __builtin_prefetch(Wp1 + kk + 128, 0, 0);
        }
        const v2f a  = *(const v2f*)(Ap  + kk);        // ds_load_b64, conflict-free
        const v2f b0 = *(const v2f*)(Wp0 + kk);        // global b64, L2-resident
        const v2f b1 = *(const v2f*)(Wp1 + kk);
        // D = A x B + C, exact fp32 path: v_wmma_f32_16x16x4_f32
        acc0 = __builtin_amdgcn_wmma_f32_16x16x4_f32(
                   false, a, false, b0, (short)0, acc0, false, false);
        acc1 = __builtin_amdgcn_wmma_f32_16x16x4_f32(
                   false, a, false, b1, (short)0, acc1, false, false);
    }

    // C/D layout: vgpr r -> M = r + 8*(lane>=16), N = lane&15
    const int mrow = (lane >> 4) << 3;   // 0 or 8
    const int nloc = lane & 15;
    #pragma unroll
    for (int r = 0; r < 8; ++r) {
        mdlds[(mrow + r) * MST + wave * 32 + nloc]      = acc0[r];
        mdlds[(mrow + r) * MST + wave * 32 + 16 + nloc] = acc1[r];
    }
    __syncthreads();

    // Finalize: each wave handles 2 rows; lane covers classes lane, lane+32, ...
    #pragma unroll
    for (int rloop = 0; rloop < 2; ++rloop) {
        const int rr  = wave * 2 + rloop;
        const int row = r0 + rr;
        const int lab = labels[row];

        float psum = 0.0f;
        float plab = 0.0f;
        #pragma unroll
        for (int c = lane; c < C_; c += 32) {
            const float md  = mdlds[rr * MST + c] + cadd[c];
            const float adj = (c == lab) ? md * (1.0f + ALPHA) : md;
            const float p   = expf(-adj) / denom[c];
            psum += p;
            if (c == lab) plab = p;
        }
        // wave32 reduction
        #pragma unroll
        for (int off = 16; off > 0; off >>= 1) {
            psum += __shfl_xor(psum, off, 32);
            plab += __shfl_xor(plab, off, 32);
        }
        if (lane == 0) {
            loss[row] = -logf(plab / (psum + EPSI) + EPSI);
        }
    }
}

// ---------------------------------------------------------------------------
extern "C" void kernel_launch(void* const* d_in, const int* in_sizes, int n_in,
                              void* d_out, int out_size, void* d_ws, size_t ws_size,
                              hipStream_t stream) {
    const float* feat   = (const float*)d_in[0];   // [2048, 256]
    const int*   labels = (const int*)  d_in[1];   // [2048]
    const float* means  = (const float*)d_in[2];   // [256, 256]
    const float* vars_  = (const float*)d_in[3];   // [256, 256]

    float* out       = (float*)d_out;
    float* loss      = out;                    // [2048]
    float* out_means = out + B_;               // [256*256]
    float* out_vars  = out_means + C_ * F_;    // [256*256]

    float* W     = (float*)d_ws;               // [256][512] = 512 KB
    float* cadd  = W + C_ * K_;                // [256]
    float* denom = cadd + C_;                  // [256]

    lgm_prep<<<C_, 256, 0, stream>>>(means, vars_, W, cadd, denom,
                                     out_means, out_vars);
    lgm_gemm<<<B_ / 16, 256, 0, stream>>>(feat, labels, W, cadd, denom, loss);
}